// GCN_66030827209227
// MI455X (gfx1250) — compile-verified
//
#include <hip/hip_runtime.h>
#include <math.h>

// ---------------------------------------------------------------------------
// GCN forward for MI455X (gfx1250, wave32).
//  h1 = x@W1 + b1            -> WMMA f32 16x16x4, 4 M-tiles per wave (B reuse)
//  a1 = sym-norm aggregate   -> edge-parallel f32 atomics (L2-resident dest)
//  h2 = relu(a1)@W2 + b2     -> WMMA f32 16x16x4 (ReLU fused into A load)
//  a2 = aggregate            -> edge-parallel f32 atomics into d_out
//  out = log_softmax(a2)     -> wave-per-row shuffle reduction, in place
//
// Roofline: aggregations move ~8-16 GB (gather + f32 atomics) -> ~0.5 ms at
// 23.3 TB/s; GEMMs are ~30 GFLOP total, negligible on the WMMA pipe even at
// full fp32, so we keep fp32 end-to-end for reference-matching numerics.
// ---------------------------------------------------------------------------

typedef float v2f __attribute__((ext_vector_type(2)));
typedef float v8f __attribute__((ext_vector_type(8)));

#define FIN 512
#define HID 256
#define CLS 64
#define MREP 4          // M-tiles per wave in the GEMMs (B-fragment reuse)

__device__ __forceinline__ void atomAddF32(float* p, float v) {
    // lowers to global_atomic_add_f32 (no CAS loop)
    unsafeAtomicAdd(p, v);
}

// ---- degree / dinv ---------------------------------------------------------
__global__ void k_fill1(float* __restrict__ p, int n) {
    int i = blockIdx.x * blockDim.x + threadIdx.x;
    if (i < n) p[i] = 1.0f;                      // self-loop contributes 1
}

__global__ void k_deg(const int* __restrict__ dst, float* __restrict__ deg, int e) {
    int i = blockIdx.x * blockDim.x + threadIdx.x;
    if (i < e) atomAddF32(&deg[dst[i]], 1.0f);
}

__global__ void k_dinv(float* __restrict__ d, int n) {
    int i = blockIdx.x * blockDim.x + threadIdx.x;
    if (i < n) d[i] = rsqrtf(fmaxf(d[i], 1.0f));
}

// ---- WMMA GEMM core: H[n,NOUT] = act(X[n,KDIM]) @ W[KDIM,NOUT] + b ---------
// One wave owns a 64x16 slab: MREP stacked 16x16 tiles sharing each B
// fragment (4 WMMAs per B load pair). blockIdx.x selects the 64-row slab,
// wave id selects the 16-col tile.
template <int KDIM, int NOUT, bool RELU>
__device__ __forceinline__ void gemm_wave(const float* __restrict__ X,
                                          const float* __restrict__ W,
                                          const float* __restrict__ bias,
                                          float* __restrict__ H, int n) {
    const int lane = threadIdx.x & 31;
    const int wave = threadIdx.x >> 5;          // N tile
    const int b0   = blockIdx.x * (16 * MREP);  // first row of slab
    const int n0   = wave << 4;
    const int mr   = lane & 15;                 // M (A) / N (B,C) within tile
    const int kb   = (lane >> 4) << 1;          // K sub-offset: 0 or 2

    const float* arow[MREP];
#pragma unroll
    for (int t = 0; t < MREP; ++t) {
        int m = b0 + t * 16 + mr;
        if (m >= n) m = n - 1;                  // clamp gathers only
        arow[t] = X + (size_t)m * KDIM;
    }
    const float* bcol = W + n0 + mr;

    v8f c[MREP];
    const float bv = bias[n0 + mr];
#pragma unroll
    for (int t = 0; t < MREP; ++t)
#pragma unroll
        for (int r = 0; r < 8; ++r) c[t][r] = bv;

#pragma unroll 2
    for (int k0 = 0; k0 < KDIM; k0 += 4) {
        v2f b;
        b.x = bcol[(size_t)(k0 + kb) * NOUT];
        b.y = bcol[(size_t)(k0 + kb + 1) * NOUT];
#pragma unroll
        for (int t = 0; t < MREP; ++t) {
            v2f a;
            if (RELU) {
                a.x = fmaxf(arow[t][k0 + kb], 0.0f);
                a.y = fmaxf(arow[t][k0 + kb + 1], 0.0f);
            } else {
                a.x = arow[t][k0 + kb];
                a.y = arow[t][k0 + kb + 1];
            }
            c[t] = __builtin_amdgcn_wmma_f32_16x16x4_f32(
                false, a, false, b, (short)0, c[t], false, false);
        }
    }

    const int rsub = (lane >> 4) << 3;          // +8 for upper half-lanes
    if (b0 + 16 * MREP <= n) {                  // fast path: slab in bounds
#pragma unroll
        for (int t = 0; t < MREP; ++t) {
            const int rbase = b0 + t * 16 + rsub;
#pragma unroll
            for (int r = 0; r < 8; ++r)
                H[(size_t)(rbase + r) * NOUT + n0 + mr] = c[t][r];
        }
    } else {                                    // tail block only
#pragma unroll
        for (int t = 0; t < MREP; ++t) {
            const int rbase = b0 + t * 16 + rsub;
#pragma unroll
            for (int r = 0; r < 8; ++r) {
                int row = rbase + r;
                if (row < n) H[(size_t)row * NOUT + n0 + mr] = c[t][r];
            }
        }
    }
}

__global__ void __launch_bounds__(512)
k_gemm1(const float* __restrict__ X, const float* __restrict__ W,
        const float* __restrict__ bias, float* __restrict__ H, int n) {
    gemm_wave<FIN, HID, false>(X, W, bias, H, n);   // 16 waves: HID/16 tiles
}

__global__ void __launch_bounds__(128)
k_gemm2(const float* __restrict__ X, const float* __restrict__ W,
        const float* __restrict__ bias, float* __restrict__ H, int n) {
    gemm_wave<HID, CLS, true>(X, W, bias, H, n);    // 4 waves: CLS/16 tiles
}

// ---- aggregation init: A[i,:] = H[i,:] * dinv[i]^2 (self loop + zero-init) -
// float4-vectorized; shift = log2(feats/4).
__global__ void k_agginit(const float* __restrict__ H, const float* __restrict__ dinv,
                          float* __restrict__ A, int total4, int shift) {
    int i = blockIdx.x * blockDim.x + threadIdx.x;
    if (i >= total4) return;
    int node = i >> shift;
    float w = dinv[node]; w *= w;
    float4 v = ((const float4*)H)[i];
    float4 o; o.x = v.x * w; o.y = v.y * w; o.z = v.z * w; o.w = v.w * w;
    ((float4*)A)[i] = o;
}

// ---- edge scatter, 256 feats: one wave per edge, 8 floats per lane ---------
__global__ void __launch_bounds__(256)
k_edge256(const int* __restrict__ src, const int* __restrict__ dst,
          const float* __restrict__ dinv, const float* __restrict__ H,
          float* __restrict__ A, int e) {
    int gw   = (blockIdx.x * blockDim.x + threadIdx.x) >> 5;
    int lane = threadIdx.x & 31;
    if (gw >= e) return;
    int s = src[gw], d = dst[gw];
    float w = dinv[s] * dinv[d];
    const float4* hp = (const float4*)(H + (size_t)s * HID);
    float*        op = A + (size_t)d * HID + (lane << 3);
    float4 v0 = hp[lane * 2 + 0];
    float4 v1 = hp[lane * 2 + 1];
    atomAddF32(op + 0, v0.x * w); atomAddF32(op + 1, v0.y * w);
    atomAddF32(op + 2, v0.z * w); atomAddF32(op + 3, v0.w * w);
    atomAddF32(op + 4, v1.x * w); atomAddF32(op + 5, v1.y * w);
    atomAddF32(op + 6, v1.z * w); atomAddF32(op + 7, v1.w * w);
}

// ---- edge scatter, 64 feats: one wave per edge, 2 floats per lane ----------
__global__ void __launch_bounds__(256)
k_edge64(const int* __restrict__ src, const int* __restrict__ dst,
         const float* __restrict__ dinv, const float* __restrict__ H,
         float* __restrict__ A, int e) {
    int gw   = (blockIdx.x * blockDim.x + threadIdx.x) >> 5;
    int lane = threadIdx.x & 31;
    if (gw >= e) return;
    int s = src[gw], d = dst[gw];
    float w = dinv[s] * dinv[d];
    float2 v = ((const float2*)(H + (size_t)s * CLS))[lane];
    float* op = A + (size_t)d * CLS + (lane << 1);
    atomAddF32(op + 0, v.x * w);
    atomAddF32(op + 1, v.y * w);
}

// ---- log_softmax over 64 classes, one wave per row, in place ---------------
__global__ void __launch_bounds__(256)
k_logsoftmax(float* __restrict__ O, int n) {
    int row  = (blockIdx.x * blockDim.x + threadIdx.x) >> 5;
    int lane = threadIdx.x & 31;
    if (row >= n) return;
    float* p = O + (size_t)row * CLS;
    float v0 = p[lane], v1 = p[lane + 32];
    float m = fmaxf(v0, v1);
#pragma unroll
    for (int off = 16; off >= 1; off >>= 1) m = fmaxf(m, __shfl_xor(m, off, 32));
    float s = __expf(v0 - m) + __expf(v1 - m);
#pragma unroll
    for (int off = 16; off >= 1; off >>= 1) s += __shfl_xor(s, off, 32);
    float l = m + __logf(s);
    p[lane]      = v0 - l;
    p[lane + 32] = v1 - l;
}

// ---------------------------------------------------------------------------
static inline size_t alignup(size_t x) { return (x + 255) & ~size_t(255); }

extern "C" void kernel_launch(void* const* d_in, const int* in_sizes, int n_in,
                              void* d_out, int out_size, void* d_ws, size_t ws_size,
                              hipStream_t stream) {
    const float* x  = (const float*)d_in[0];
    const int*   ei = (const int*)  d_in[1];
    const float* W1 = (const float*)d_in[2];
    const float* b1 = (const float*)d_in[3];
    const float* W2 = (const float*)d_in[4];
    const float* b2 = (const float*)d_in[5];
    float* out = (float*)d_out;

    const int N = in_sizes[0] / FIN;
    const int E = in_sizes[1] / 2;
    const int* src = ei;
    const int* dst = ei + E;

    char* ws = (char*)d_ws;
    size_t o0 = 0;                                   // dinv  [N]
    size_t o1 = alignup(o0 + (size_t)N * 4);         // h1    [N,HID]
    size_t o2 = alignup(o1 + (size_t)N * HID * 4);   // agg1  [N,HID]
    size_t o3 = alignup(o2 + (size_t)N * HID * 4);   // h2    [N,CLS]
    float* dinv = (float*)(ws + o0);
    float* h1   = (float*)(ws + o1);
    float* agg1 = (float*)(ws + o2);
    float* h2   = (float*)(ws + o3);

    const int slabs = (N + 16 * MREP - 1) / (16 * MREP);   // 64-row GEMM slabs
    const int eb    = (E + 7) / 8;                         // wave/edge, 8 w/blk

    // degrees -> dinv
    k_fill1<<<(N + 255) / 256, 256, 0, stream>>>(dinv, N);
    k_deg  <<<(E + 255) / 256, 256, 0, stream>>>(dst, dinv, E);
    k_dinv <<<(N + 255) / 256, 256, 0, stream>>>(dinv, N);

    // layer 1
    k_gemm1<<<slabs, 512, 0, stream>>>(x, W1, b1, h1, N);
    k_agginit<<<(N * (HID / 4) + 255) / 256, 256, 0, stream>>>(
        h1, dinv, agg1, N * (HID / 4), 6);
    k_edge256<<<eb, 256, 0, stream>>>(src, dst, dinv, h1, agg1, E);

    // layer 2 (ReLU fused into GEMM2 A-load)
    k_gemm2<<<slabs, 128, 0, stream>>>(agg1, W2, b2, h2, N);
    k_agginit<<<(N * (CLS / 4) + 255) / 256, 256, 0, stream>>>(
        h2, dinv, out, N * (CLS / 4), 4);
    k_edge64<<<eb, 256, 0, stream>>>(src, dst, dinv, h2, out, E);

    // log-softmax in place on d_out
    k_logsoftmax<<<(N + 7) / 8, 256, 0, stream>>>(out, N);
}